// DecNP_41188736369124
// MI455X (gfx1250) — compile-verified
//
#include <hip/hip_runtime.h>
#include <math.h>

typedef float v2f __attribute__((ext_vector_type(2)));
typedef float v8f __attribute__((ext_vector_type(8)));

// ---------------------------------------------------------------------------
// top-3 (ascending) insert, branchless
// ---------------------------------------------------------------------------
__device__ __forceinline__ void top3_insert(float v, int mi,
                                            float& d0, float& d1, float& d2,
                                            int& i0, int& i1, int& i2) {
    bool lt2 = v < d2, lt1 = v < d1, lt0 = v < d0;
    float nd2 = lt1 ? d1 : (lt2 ? v : d2);
    int   ni2 = lt1 ? i1 : (lt2 ? mi : i2);
    float nd1 = lt0 ? d0 : (lt1 ? v : d1);
    int   ni1 = lt0 ? i0 : (lt1 ? mi : i1);
    d0 = lt0 ? v : d0;  i0 = lt0 ? mi : i0;
    d1 = nd1;           i1 = ni1;
    d2 = nd2;           i2 = ni2;
}

// ---------------------------------------------------------------------------
// 3-NN search via V_WMMA_F32_16X16X4_F32.
// A (16x4, coarse m): row m = (-2cx, -2cy, -2cz, |c|^2)
// B (4x16, fine n)  : col n = ( fx,   fy,   fz,  1   )
// D[m][n] = -2 c_m . f_n + |c_m|^2   (add per-lane |f_n|^2 at the end; it is
// constant per column so argmin is unaffected).
// One wave handles 16 fine points; 8 waves per 256-thread block.
// ---------------------------------------------------------------------------
__global__ __launch_bounds__(256) void knn3_wmma(
    const float* __restrict__ xyzF,   // [B, N, 3]
    const float* __restrict__ xyzC,   // [B, S, 3]
    int*   __restrict__ idxOut,       // [B, N, 3]
    float* __restrict__ wOut,         // [B, N, 3]
    int Bn, int N, int S)
{
    const int lane = threadIdx.x & 31;
    const int wave = threadIdx.x >> 5;
    const int tile = blockIdx.x * 8 + wave;          // wave-uniform
    const int tilesPerBatch = N >> 4;
    if (tile >= Bn * tilesPerBatch) return;          // whole wave exits together
    const int b  = tile / tilesPerBatch;
    const int n0 = (tile % tilesPerBatch) << 4;

    const int nl = lane & 15;
    const int hi = lane >> 4;                        // 0: rows 0-7 / K=0,1 ; 1: rows 8-15 / K=2,3

    // ---- B operand (fine points), constant over the coarse loop ----
    const float* fp = xyzF + ((size_t)b * N + (n0 + nl)) * 3;
    float fx = fp[0], fy = fp[1], fz = fp[2];
    float fsq = fx * fx + fy * fy + fz * fz;
    v2f Bm;
    Bm.x = hi ? fz   : fx;    // VGPR0: K=0 (lanes 0-15), K=2 (lanes 16-31)
    Bm.y = hi ? 1.0f : fy;    // VGPR1: K=1 (lanes 0-15), K=3 (lanes 16-31)

    float d0 = 1e30f, d1 = 1e30f, d2 = 1e30f;
    int   i0 = 0,     i1 = 0,     i2 = 0;

    const float* cbase = xyzC + (size_t)b * S * 3;
    for (int t = 0; t < S; t += 16) {
        const float* cp = cbase + (size_t)(t + nl) * 3;
        float cx = cp[0], cy = cp[1], cz = cp[2];
        float csq = cx * cx + cy * cy + cz * cz;
        v2f Am;
        Am.x = hi ? (-2.0f * cz) : (-2.0f * cx);
        Am.y = hi ? csq          : (-2.0f * cy);

        v8f acc = {};
        acc = __builtin_amdgcn_wmma_f32_16x16x4_f32(
            /*neg_a=*/false, Am, /*neg_b=*/false, Bm,
            /*c_mod=*/(short)0, acc, /*reuse_a=*/false, /*reuse_b=*/false);

        // lane holds column n = nl; VGPR r holds row m = r + hi*8
#pragma unroll
        for (int r = 0; r < 8; ++r) {
            float val = acc[r];
            int   mi  = t + r + (hi << 3);
            top3_insert(val, mi, d0, d1, d2, i0, i1, i2);
        }
    }

    // ---- merge the two half-lane top-3 lists (lane n <-> lane n+16) ----
    float e0 = __shfl_xor(d0, 16, 32);
    float e1 = __shfl_xor(d1, 16, 32);
    float e2 = __shfl_xor(d2, 16, 32);
    int   j0 = __shfl_xor(i0, 16, 32);
    int   j1 = __shfl_xor(i1, 16, 32);
    int   j2 = __shfl_xor(i2, 16, 32);
    top3_insert(e0, j0, d0, d1, d2, i0, i1, i2);
    top3_insert(e1, j1, d0, d1, d2, i0, i1, i2);
    top3_insert(e2, j2, d0, d1, d2, i0, i1, i2);

    // ---- inverse-distance weights ----
    float dd0 = d0 + fsq, dd1 = d1 + fsq, dd2 = d2 + fsq;
    float w0 = 1.0f / (dd0 + 1e-8f);
    float w1 = 1.0f / (dd1 + 1e-8f);
    float w2 = 1.0f / (dd2 + 1e-8f);
    float wsum = w0 + w1 + w2;
    w0 /= wsum; w1 /= wsum; w2 /= wsum;

    if (hi == 0) {
        size_t o = ((size_t)b * N + n0 + nl) * 3;
        idxOut[o + 0] = i0; idxOut[o + 1] = i1; idxOut[o + 2] = i2;
        wOut[o + 0]   = w0; wOut[o + 1]   = w1; wOut[o + 2]   = w2;
    }
}

// ---------------------------------------------------------------------------
// Pass 1: per-block partial (sum, sumsq) of interpolated features.
// Element i = (b*C2 + c)*N + n ; each block owns 8192 elements.
// ---------------------------------------------------------------------------
__global__ __launch_bounds__(256) void interp_stats(
    const float* __restrict__ pts2,   // [B, C2, S]
    const int*   __restrict__ idx,    // [B, N, 3]
    const float* __restrict__ w,      // [B, N, 3]
    float* __restrict__ partials,     // [gridDim.x * 2]
    int N, int Nshift, int S, int C2, long long total)
{
    __shared__ float ssum[256], ssq[256];
    const int tid = threadIdx.x;
    long long base = (long long)blockIdx.x * 8192 + tid;
    float s = 0.0f, q = 0.0f;
#pragma unroll 4
    for (int it = 0; it < 32; ++it) {
        long long i = base + (long long)it * 256;
        if (i < total) {
            int n  = (int)(i & (N - 1));
            int bc = (int)(i >> Nshift);          // b*C2 + c
            int b  = bc / C2;
            size_t row = (size_t)bc * S;
            size_t o   = ((size_t)b * N + n) * 3;
            float v = w[o + 0] * pts2[row + idx[o + 0]]
                    + w[o + 1] * pts2[row + idx[o + 1]]
                    + w[o + 2] * pts2[row + idx[o + 2]];
            s += v; q += v * v;
        }
    }
    ssum[tid] = s; ssq[tid] = q;
    __syncthreads();
    for (int off = 128; off > 0; off >>= 1) {
        if (tid < off) { ssum[tid] += ssum[tid + off]; ssq[tid] += ssq[tid + off]; }
        __syncthreads();
    }
    if (tid == 0) {
        partials[2 * blockIdx.x + 0] = ssum[0];
        partials[2 * blockIdx.x + 1] = ssq[0];
    }
}

// ---------------------------------------------------------------------------
// Pass 2: single-block deterministic reduction -> (mean, std ddof=1)
// ---------------------------------------------------------------------------
__global__ __launch_bounds__(256) void finalize_stats(
    const float* __restrict__ partials, int nParts,
    float* __restrict__ stats, float cnt)
{
    __shared__ float ssum[256], ssq[256];
    const int tid = threadIdx.x;
    float s = 0.0f, q = 0.0f;
    for (int i = tid; i < nParts; i += 256) {
        s += partials[2 * i + 0];
        q += partials[2 * i + 1];
    }
    ssum[tid] = s; ssq[tid] = q;
    __syncthreads();
    for (int off = 128; off > 0; off >>= 1) {
        if (tid < off) { ssum[tid] += ssum[tid + off]; ssq[tid] += ssq[tid + off]; }
        __syncthreads();
    }
    if (tid == 0) {
        float m   = ssum[0] / cnt;
        float var = (ssq[0] - cnt * m * m) / (cnt - 1.0f);
        var = var < 0.0f ? 0.0f : var;
        stats[0] = m;
        stats[1] = sqrtf(var);
    }
}

// ---------------------------------------------------------------------------
// Write [B, C1+C2, N]: copy skip features, recompute + normalize interp.
// ---------------------------------------------------------------------------
__global__ __launch_bounds__(256) void write_out(
    const float* __restrict__ pts1,   // [B, C1, N]
    const float* __restrict__ pts2,   // [B, C2, S]
    const int*   __restrict__ idx,    // [B, N, 3]
    const float* __restrict__ w,      // [B, N, 3]
    const float* __restrict__ stats,  // [2]: m, s
    float* __restrict__ out,          // [B, C1+C2, N]
    int N, int Nshift, int S, int C1, int C2, long long total)
{
    long long i = (long long)blockIdx.x * 256 + threadIdx.x;
    if (i >= total) return;
    const int Ctot = C1 + C2;
    int n  = (int)(i & (N - 1));
    int bc = (int)(i >> Nshift);     // b*Ctot + c
    int c  = bc % Ctot;
    int b  = bc / Ctot;
    float v;
    if (c < C1) {
        v = pts1[((size_t)b * C1 + c) * N + n];
    } else {
        int cc = c - C1;
        size_t row = ((size_t)b * C2 + cc) * S;
        size_t o   = ((size_t)b * N + n) * 3;
        float val = w[o + 0] * pts2[row + idx[o + 0]]
                  + w[o + 1] * pts2[row + idx[o + 1]]
                  + w[o + 2] * pts2[row + idx[o + 2]];
        v = (val - stats[0]) / (stats[1] + 1e-5f);
    }
    out[i] = v;
}

// ---------------------------------------------------------------------------
// Host-side orchestration
// ---------------------------------------------------------------------------
extern "C" void kernel_launch(void* const* d_in, const int* in_sizes, int n_in,
                              void* d_out, int out_size, void* d_ws, size_t ws_size,
                              hipStream_t stream) {
    constexpr int B  = 8;
    constexpr int N1 = 2048, S1 = 512,  C1a = 256, C2a = 512;   // level 1
    constexpr int N2 = 8192, S2 = 2048, C1b = 128, C2b = 768;   // level 2
    constexpr int Cout1 = C1a + C2a;   // 768
    // inputs: xyz0, xyz1, xyz2, x0, x1, x2
    const float* xyz0 = (const float*)d_in[0];
    const float* xyz1 = (const float*)d_in[1];
    const float* xyz2 = (const float*)d_in[2];
    const float* x0   = (const float*)d_in[3];
    const float* x1   = (const float*)d_in[4];
    const float* x2   = (const float*)d_in[5];
    float* out = (float*)d_out;

    // workspace layout
    char* ws = (char*)d_ws;
    int*   idx1     = (int*)  (ws + 0);                       // 196,608 B
    float* w1       = (float*)(ws + 196608);                  // 196,608 B
    int*   idx2     = (int*)  (ws + 393216);                  // 786,432 B
    float* w2       = (float*)(ws + 1179648);                 // 786,432 B
    float* partials = (float*)(ws + 1966080);                 //  49,152 B
    float* stats1   = (float*)(ws + 2015232);                 //      16 B
    float* stats2   = (float*)(ws + 2015248);                 //      16 B
    float* out1     = (float*)(ws + 2097152);                 // 50,331,648 B  [B,768,N1]

    // ---------------- Level 1: fine=xyz1, coarse=xyz2, feats=x2 ----------------
    {
        int tiles = B * (N1 >> 4);                            // 1024
        knn3_wmma<<<(tiles + 7) / 8, 256, 0, stream>>>(xyz1, xyz2, idx1, w1, B, N1, S1);

        long long total = (long long)B * C2a * N1;            // 8,388,608
        int nBlocks = (int)((total + 8191) / 8192);           // 1024
        interp_stats<<<nBlocks, 256, 0, stream>>>(x2, idx1, w1, partials,
                                                  N1, 11, S1, C2a, total);
        finalize_stats<<<1, 256, 0, stream>>>(partials, nBlocks, stats1, (float)total);

        long long totW = (long long)B * Cout1 * N1;           // 12,582,912
        write_out<<<(int)((totW + 255) / 256), 256, 0, stream>>>(
            x1, x2, idx1, w1, stats1, out1, N1, 11, S1, C1a, C2a, totW);
    }

    // ---------------- Level 2: fine=xyz0, coarse=xyz1, feats=out1 --------------
    {
        int tiles = B * (N2 >> 4);                            // 4096
        knn3_wmma<<<(tiles + 7) / 8, 256, 0, stream>>>(xyz0, xyz1, idx2, w2, B, N2, S2);

        long long total = (long long)B * C2b * N2;            // 50,331,648
        int nBlocks = (int)((total + 8191) / 8192);           // 6144
        interp_stats<<<nBlocks, 256, 0, stream>>>(out1, idx2, w2, partials,
                                                  N2, 13, S2, C2b, total);
        finalize_stats<<<1, 256, 0, stream>>>(partials, nBlocks, stats2, (float)total);

        long long totW = (long long)B * (C1b + C2b) * N2;     // 58,720,256
        write_out<<<(int)((totW + 255) / 256), 256, 0, stream>>>(
            x0, out1, idx2, w2, stats2, out, N2, 13, S2, C1b, C2b, totW);
    }
}